// NonGalerkinLoss3Eig_87041807221109
// MI455X (gfx1250) — compile-verified
//
#include <hip/hip_runtime.h>

// ---------------------------------------------------------------------------
// NonGalerkinLoss3Eig for MI455X (gfx1250, wave32).
// Per sample: symmetric eigensolve (parallel cyclic Jacobi in LDS, padded
// stride N+1 to kill bank conflicts), matrix power via binary exponentiation
// with V_WMMA_F32_16X16X4_F32 matmuls, A@V with WMMA, scalar reductions,
// atomicAdd into the single float output.
// ---------------------------------------------------------------------------

typedef float v2f __attribute__((ext_vector_type(2)));
typedef float v8f __attribute__((ext_vector_type(8)));

#define NT      256   // 8 wave32 per block
#define NSWEEP  10    // Jacobi sweeps (fp32 convergence for n<=64)
#define KSEL    25    // top-k eigenvalues (min(dim,25) == 25 for dim 32,64)
#define MCONSTM1 19.0f

__global__ void zero_out_kernel(float* out) { out[0] = 0.0f; }

// C = A @ B, all DIM x DIM row-major fp32 in LDS with padded row stride LD.
// Each wave computes 16x16 tiles of C; K accumulated in chunks of 4 via
// V_WMMA_F32_16X16X4_F32 (fp32 matrix core path on CDNA5).
// VGPR layouts per CDNA5 ISA 7.12.2:
//   A 16x4:  lanes 0-15 rows, v[0]=K0,v[1]=K1; lanes 16-31: v[0]=K2,v[1]=K3
//   B 4x16:  v[0]: K0 (lanes0-15) / K2 (lanes16-31); v[1]: K1 / K3
//   C 16x16: v[r]: M=r (lanes0-15) / M=r+8 (lanes16-31), N=lane%16
template<int DIM>
__device__ inline void mm_wmma(const float* A, const float* B, float* C,
                               int lane, int wave, int nwaves) {
    constexpr int LD  = DIM + 1;
    constexpr int T16 = DIM / 16;
    constexpr int KC  = DIM / 4;
    const int half = lane >> 4;   // 0 or 1
    const int l    = lane & 15;
    for (int tile = wave; tile < T16 * T16; tile += nwaves) {
        const int ti = tile / T16;
        const int tj = tile % T16;
        v8f acc = {0.f, 0.f, 0.f, 0.f, 0.f, 0.f, 0.f, 0.f};
        const int arow = ti * 16 + l;
        const int bcol = tj * 16 + l;
        for (int kc = 0; kc < KC; ++kc) {
            const int k0 = kc * 4 + 2 * half;
            v2f a, b;
            a.x = A[arow * LD + k0];
            a.y = A[arow * LD + k0 + 1];
            b.x = B[k0 * LD + bcol];
            b.y = B[(k0 + 1) * LD + bcol];
            acc = __builtin_amdgcn_wmma_f32_16x16x4_f32(
                false, a, false, b, (short)0, acc, false, false);
        }
        for (int r = 0; r < 8; ++r) {
            C[(ti * 16 + half * 8 + r) * LD + tj * 16 + l] = acc[r];
        }
    }
}

template<int DIM>
__global__ __launch_bounds__(NT)
void loss_kernel(const float* __restrict__ inp,
                 const float* __restrict__ outp,
                 float* __restrict__ out) {
    constexpr int N      = DIM;
    constexpr int LD     = N + 1;          // padded LDS row stride
    constexpr int NN     = N * N;
    constexpr int NNP    = N * LD;         // padded matrix footprint (floats)
    constexpr int NPAIR  = N / 2;
    constexpr int NWAVES = NT / 32;

    extern __shared__ float smem[];
    float* bufS   = smem;             // sym matrix -> Jacobi workspace / matpow temp
    float* bufV   = smem + 1 * NNP;   // eigenvectors
    float* bufPhi = smem + 2 * NNP;   // phi -> matpow base
    float* bufR   = smem + 3 * NNP;   // matpow result

    __shared__ float cA[32], sA[32];
    __shared__ int   pA[32], qA[32];
    __shared__ float wEig[64];
    __shared__ int   used[64];
    __shared__ int   sel[KSEL];
    __shared__ float lamSel[KSEL];
    __shared__ float aOnes[64], pOnes[64];
    __shared__ float contrib[KSEL];

    const int tid  = threadIdx.x;
    const int lane = tid & 31;
    const int wave = tid >> 5;

    const int s = blockIdx.x;
    const float* phiRow = inp  + (size_t)s * 4097;
    const float* psiRow = outp + (size_t)s * 4096;
    const int m = (int)phiRow[4096];

    // ---- load phi; build sym S; V = I; R = I -------------------------------
    for (int idx = tid; idx < NN; idx += NT) {
        const int i = idx / N, j = idx % N;
        bufPhi[i * LD + j] = phiRow[idx];
    }
    __syncthreads();
    for (int idx = tid; idx < NN; idx += NT) {
        const int i = idx / N, j = idx % N;
        bufS[i * LD + j] = (i >= j) ? bufPhi[i * LD + j] : bufPhi[j * LD + i];
        bufV[i * LD + j] = (i == j) ? 1.f : 0.f;
        bufR[i * LD + j] = (i == j) ? 1.f : 0.f;
    }
    __syncthreads();

    // ---- parallel cyclic Jacobi eigensolve ---------------------------------
    for (int sweep = 0; sweep < NSWEEP; ++sweep) {
        for (int r = 0; r < N - 1; ++r) {
            if (tid < NPAIR) {
                int p, q;
                if (tid == 0) { p = N - 1; q = r % (N - 1); }
                else {
                    p = (r + tid) % (N - 1);
                    q = (r - tid + (N - 1)) % (N - 1);
                }
                const float app = bufS[p * LD + p];
                const float aqq = bufS[q * LD + q];
                const float apq = bufS[p * LD + q];
                float c = 1.f, sn = 0.f;
                if (fabsf(apq) > 1e-30f) {
                    const float tau = (aqq - app) / (2.f * apq);
                    const float tt  = ((tau >= 0.f) ? 1.f : -1.f) /
                                      (fabsf(tau) + sqrtf(1.f + tau * tau));
                    c  = 1.f / sqrtf(1.f + tt * tt);
                    sn = tt * c;
                }
                pA[tid] = p; qA[tid] = q; cA[tid] = c; sA[tid] = sn;
            }
            __syncthreads();
            // column pass: S <- S*J ; V <- V*J (disjoint pairs -> race-free;
            // padded stride makes bank = (row+p) % 64 -> conflict-free)
            for (int idx = tid; idx < NPAIR * N; idx += NT) {
                const int i = idx / N, row = idx % N;
                const int p = pA[i], q = qA[i];
                const float c = cA[i], sn = sA[i];
                const float sp = bufS[row * LD + p], sq = bufS[row * LD + q];
                bufS[row * LD + p] = c * sp - sn * sq;
                bufS[row * LD + q] = sn * sp + c * sq;
                const float vp = bufV[row * LD + p], vq = bufV[row * LD + q];
                bufV[row * LD + p] = c * vp - sn * vq;
                bufV[row * LD + q] = sn * vp + c * vq;
            }
            __syncthreads();
            // row pass: S <- J^T * S
            for (int idx = tid; idx < NPAIR * N; idx += NT) {
                const int i = idx / N, col = idx % N;
                const int p = pA[i], q = qA[i];
                const float c = cA[i], sn = sA[i];
                const float sp = bufS[p * LD + col], sq = bufS[q * LD + col];
                bufS[p * LD + col] = c * sp - sn * sq;
                bufS[q * LD + col] = sn * sp + c * sq;
            }
            __syncthreads();
        }
    }

    // ---- eigenvalues; top-25 by |lambda| -----------------------------------
    if (tid < N) {
        wEig[tid] = bufS[tid * LD + tid];
        used[tid] = 0;
    }
    __syncthreads();
    if (tid == 0) {
        for (int jj = 0; jj < KSEL; ++jj) {
            float best = -1.f; int bi = 0;
            for (int j = 0; j < N; ++j) {
                if (!used[j]) {
                    const float a = fabsf(wEig[j]);
                    if (a > best) { best = a; bi = j; }
                }
            }
            used[bi] = 1;
            sel[jj] = bi;
            lamSel[jj] = wEig[bi];
        }
    }
    __syncthreads();

    // ---- phi^m via binary exponentiation (WMMA matmuls) --------------------
    float* Rp = bufR;     // running result (starts as I)
    float* Bp = bufPhi;   // running base   (starts as phi)
    float* Tp = bufS;     // temp (S is free now)
    int mm = m;
    for (int bit = 0; bit < 3; ++bit) {
        if (mm & 1) {
            mm_wmma<DIM>(Rp, Bp, Tp, lane, wave, NWAVES);
            __syncthreads();
            float* t = Rp; Rp = Tp; Tp = t;
        }
        if (bit < 2) {
            mm_wmma<DIM>(Bp, Bp, Tp, lane, wave, NWAVES);
            __syncthreads();
            float* t = Bp; Bp = Tp; Tp = t;
        }
        mm >>= 1;
    }
    // Rp = phi^m ; Bp, Tp free

    // ---- A = psi - phi^m  (psi rows are 16B aligned -> float4 loads) -------
    {
        const float4* psi4 = (const float4*)psiRow;
        constexpr int NQ = N / 4;
        for (int idx4 = tid; idx4 < NN / 4; idx4 += NT) {
            const int i  = idx4 / NQ;
            const int j0 = (idx4 % NQ) * 4;
            const float4 pv = psi4[idx4];
            float* dst = Bp + i * LD + j0;
            const float* pm = Rp + i * LD + j0;
            dst[0] = pv.x - pm[0];
            dst[1] = pv.y - pm[1];
            dst[2] = pv.z - pm[2];
            dst[3] = pv.w - pm[3];
        }
    }
    __syncthreads();

    // row sums for the rank-one term
    if (tid < N) {
        float sa = 0.f, sp = 0.f;
        for (int j = 0; j < N; ++j) { sa += Bp[tid * LD + j]; sp += Rp[tid * LD + j]; }
        aOnes[tid] = sa; pOnes[tid] = sp;
    }
    __syncthreads();

    // ---- AV = A @ V (WMMA) -------------------------------------------------
    mm_wmma<DIM>(Bp, bufV, Tp, lane, wave, NWAVES);
    __syncthreads();

    // ---- eigen-term contributions ------------------------------------------
    if (tid < KSEL) {
        const int cidx = sel[tid];
        float s2 = 0.f;
        for (int r2 = 0; r2 < N; ++r2) {
            const float v = Tp[r2 * LD + cidx];
            s2 += v * v;
        }
        // lam^m (binary pow, matches reference incl. negative lam)
        float resp = 1.f, base = lamSel[tid];
        int mb = m;
        for (int b = 0; b < 3; ++b) { if (mb & 1) resp *= base; base *= base; mb >>= 1; }
        float den = 1.f + MCONSTM1 * fabsf(1.f - resp);
        den = den * den * (float)(KSEL + 1);
        contrib[tid] = MCONSTM1 * s2 / den;
    }
    __syncthreads();

    if (tid == 0) {
        float tEig = 0.f;
        for (int j = 0; j < KSEL; ++j) tEig += contrib[j];
        float sna = 0.f, snp = 0.f;
        for (int i = 0; i < N; ++i) {
            sna += aOnes[i] * aOnes[i];
            const float d = 1.f - pOnes[i];
            snp += d * d;
        }
        const float num1 = MCONSTM1 * sna;
        const float den1 = ((float)N + MCONSTM1 * snp) * (float)(KSEL + 1);
        atomicAdd(out, tEig + num1 / den1);
    }
}

extern "C" void kernel_launch(void* const* d_in, const int* in_sizes, int n_in,
                              void* d_out, int out_size, void* d_ws, size_t ws_size,
                              hipStream_t stream) {
    const float* inp  = (const float*)d_in[0];   // (B, 4097): phi | m
    const float* outp = (const float*)d_in[1];   // (B, 4096): psi
    float* out = (float*)d_out;                  // single float loss

    const int B = in_sizes[0] / 4097;            // 4096 samples

    zero_out_kernel<<<1, 1, 0, stream>>>(out);

    // dynamic LDS: 4 matrices of DIM*(DIM+1) floats (padded stride)
    loss_kernel<32><<<B, NT, 4 * 32 * 33 * (int)sizeof(float), stream>>>(inp, outp, out);
    loss_kernel<64><<<B, NT, 4 * 64 * 65 * (int)sizeof(float), stream>>>(inp, outp, out);
}